// symmetry_module_23278722744737
// MI455X (gfx1250) — compile-verified
//
#include <hip/hip_runtime.h>
#include <float.h>

#define T_FRAMES 300
#define V_JOINTS 25
#define TILE_T   64
#define THREADS  256

// ---- gfx1250 async global->LDS path (guarded so compile never breaks) ----
#if defined(__AMDGCN__) && __has_builtin(__builtin_amdgcn_global_load_async_to_lds_b128)
#define HAVE_ASYNC_LDS 1
typedef int v4i __attribute__((vector_size(4 * sizeof(int))));
typedef __attribute__((address_space(1))) v4i* gas1_v4p;   // global src (B128)
typedef __attribute__((address_space(3))) v4i* las3_v4p;   // LDS dst   (B128)
#else
#define HAVE_ASYNC_LDS 0
#endif

__global__ __launch_bounds__(THREADS)
void angle_kernel(const float* __restrict__ x, float* __restrict__ out)
{
    // LDS mirror of one (n, 64-frame) tile, all 3 coordinate planes:
    // lds[c][t_local][v] -> c*TILE_T*25 + t_local*25 + v
    __shared__ float lds[3 * TILE_T * V_JOINTS];

    const int tilesPerN = (T_FRAMES + TILE_T - 1) / TILE_T;  // 5
    const int blk  = blockIdx.x;
    const int n    = blk / tilesPerN;
    const int tile = blk - n * tilesPerN;
    const int t0   = tile * TILE_T;
    const int tf   = (T_FRAMES - t0) < TILE_T ? (T_FRAMES - t0) : TILE_T; // 64 or 44
    const int tid  = threadIdx.x;

    // ---- stage tile into LDS: 3 planes, each tf*25 floats, contiguous, 16B aligned ----
    const int chunks = (tf * V_JOINTS) >> 2;   // float4 chunks per plane (1600/4 or 1100/4)
    for (int c = 0; c < 3; ++c) {
        const float* gsrc = x + (((size_t)n * 3 + c) * T_FRAMES + t0) * V_JOINTS;
        float* lb = &lds[c * TILE_T * V_JOINTS];
#if HAVE_ASYNC_LDS
        for (int i = tid; i < chunks; i += THREADS) {
            __builtin_amdgcn_global_load_async_to_lds_b128(
                (gas1_v4p)(gsrc + 4 * i), (las3_v4p)(lb + 4 * i), 0, 0);
        }
#else
        for (int i = tid; i < chunks; i += THREADS) {
            *(float4*)(lb + 4 * i) = *(const float4*)(gsrc + 4 * i);
        }
#endif
    }
#if HAVE_ASYNC_LDS
#if __has_builtin(__builtin_amdgcn_s_wait_asynccnt)
    __builtin_amdgcn_s_wait_asynccnt(0);
#else
    asm volatile("s_wait_asynccnt 0" ::: "memory");
#endif
#endif
    __syncthreads();

    // ---- compute: angle[n,t,v] = <h/|h|, vec_v/|vec_v|>, h = x16-x12, vec_v = x_v-x16 ----
    const int nOut = tf * V_JOINTS;
    float* outBase = out + ((size_t)n * T_FRAMES + t0) * V_JOINTS;
    for (int j = tid; j < nOut; j += THREADS) {
        const int tl = j / V_JOINTS;
        const int v  = j - tl * V_JOINTS;
        const float* p0 = &lds[0 * TILE_T * V_JOINTS + tl * V_JOINTS];
        const float* p1 = &lds[1 * TILE_T * V_JOINTS + tl * V_JOINTS];
        const float* p2 = &lds[2 * TILE_T * V_JOINTS + tl * V_JOINTS];

        const float hx = p0[16] - p0[12];
        const float hy = p1[16] - p1[12];
        const float hz = p2[16] - p2[12];
        const float rh = rsqrtf(hx * hx + hy * hy + hz * hz);

        float vx, vy, vz;
        if (v == 16) { vx = hx; vy = hy; vz = hz; }            // vec[16] = hip_vec
        else {
            vx = p0[v] - p0[16];
            vy = p1[v] - p1[16];
            vz = p2[v] - p2[16];
        }
        const float rv = rsqrtf(vx * vx + vy * vy + vz * vz);

        float a = (hx * vx + hy * vy + hz * vz) * (rh * rv);
        if (!(a == a)) a = 0.0f;                               // nan_to_num: NaN -> 0
        a = fminf(fmaxf(a, -FLT_MAX), FLT_MAX);                // nan_to_num: +/-inf clamp
        if (a < 0.0f) a += 1.0f;                               // where(angle<0, angle+1)

        __builtin_nontemporal_store(a, outBase + j);           // write-once output
    }
}

extern "C" void kernel_launch(void* const* d_in, const int* in_sizes, int n_in,
                              void* d_out, int out_size, void* d_ws, size_t ws_size,
                              hipStream_t stream)
{
    const float* x = (const float*)d_in[0];
    float* out     = (float*)d_out;
    const int N    = in_sizes[0] / (3 * T_FRAMES * V_JOINTS);   // 1024
    const int tilesPerN = (T_FRAMES + TILE_T - 1) / TILE_T;     // 5
    dim3 grid((unsigned)(N * tilesPerN));
    angle_kernel<<<grid, THREADS, 0, stream>>>(x, out);
}